// TypeSpecificEncoder_74929999446089
// MI455X (gfx1250) — compile-verified
//
#include <hip/hip_runtime.h>

typedef __attribute__((ext_vector_type(2))) float v2f;
typedef __attribute__((ext_vector_type(8))) float v8f;

#define NUM_HEADS 4
#define WORD_DIM  100
#define DPAD      112     // 7 tiles of 16
#define NWORDS    39
#define WTOT      40
#define LEAKY     0.3f
#define BTE       (8 * 30 * 20)

__global__ __launch_bounds__(256)
void tse_kernel(const float* __restrict__ x,
                const float* __restrict__ w_att,
                const float* __restrict__ b_att,
                float* __restrict__ out)
{
    // Padded LDS staging: rows 0..39 = x words (row 0 = type_emb), row 40 = zero.
    __shared__ float xs[41 * DPAD];
    __shared__ float wb[2 * NUM_HEADS * DPAD];   // w_att then b_att, col-padded
    __shared__ float att[16 * 40];               // WMMA A: rows>=4 and col 39 stay zero

    const int tid = threadIdx.x;
    const int ent = blockIdx.x;                  // flattened (b,t,e)

    // ---- bulk load: 40x100 fp32 slice = exactly 1000 float4, none straddle a row ----
    const float4* xg4 = (const float4*)(x + (size_t)ent * (WTOT * WORD_DIM));
    bool myany = false;
    for (int j = tid; j < 1000; j += 256) {
        float4 v = xg4[j];                        // global_load_b128, fully coalesced
        int row = j / 25;                         // 25 float4 per 100-float row
        int c4  = j - row * 25;
        *(float4*)&xs[row * DPAD + c4 * 4] = v;   // ds_store_b128 (16B aligned: 448B rows)
        myany = myany | (v.x != 0.f) | (v.y != 0.f) | (v.z != 0.f) | (v.w != 0.f);
    }
    // zero-fill pads: cols 100..111 of rows 0..40 (41*3 f4) + row 40 cols 0..99 (25 f4)
    {
        const float4 z = {0.f, 0.f, 0.f, 0.f};
        for (int j = tid; j < 148; j += 256) {
            int adw;
            if (j < 123) { int r = j / 3; adw = r * DPAD + WORD_DIM + (j - r * 3) * 4; }
            else         { adw = WTOT * DPAD + (j - 123) * 4; }
            *(float4*)&xs[adw] = z;
        }
        for (int j = tid; j < 160; j += 256)      // att: 640 floats = 160 f4
            *(float4*)&att[j * 4] = z;
    }
    // w_att / b_att: 4x100 each = 100 float4 each (pad cols never read)
    {
        const float4* wg4 = (const float4*)w_att;
        const float4* bg4 = (const float4*)b_att;
        for (int j = tid; j < 100; j += 256) {
            int h = j / 25, c4 = j - h * 25;
            *(float4*)&wb[h * DPAD + c4 * 4]                  = wg4[j];
            *(float4*)&wb[NUM_HEADS * DPAD + h * DPAD + c4 * 4] = bg4[j];
        }
    }
    // barrier + block-wide OR of the nonzero mask in one op
    const float flagf = __syncthreads_or((int)myany) ? 1.f : 0.f;

    // ---- scores[h,n] = sum_d type_emb[d] * leaky(words[n,d]*w[h,d] + b[h,d]); exp ----
    if (tid < NUM_HEADS * NWORDS) {
        int n = tid >> 2;                        // 0..38
        int h = tid & 3;                         // 0..3
        const float4* wr = (const float4*)&wb[h * DPAD];
        const float4* br = (const float4*)&wb[NUM_HEADS * DPAD + h * DPAD];
        const float4* tr = (const float4*)&xs[0];
        const float4* wd = (const float4*)&xs[(n + 1) * DPAD];
        float s = 0.f;
        #pragma unroll 5
        for (int q = 0; q < 25; ++q) {           // ds_load_b128 x4 per step
            float4 w4 = wd[q], a4 = wr[q], b4 = br[q], t4 = tr[q];
            float p;
            p = fmaf(w4.x, a4.x, b4.x); p = (p >= 0.f) ? p : LEAKY * p; s = fmaf(t4.x, p, s);
            p = fmaf(w4.y, a4.y, b4.y); p = (p >= 0.f) ? p : LEAKY * p; s = fmaf(t4.y, p, s);
            p = fmaf(w4.z, a4.z, b4.z); p = (p >= 0.f) ? p : LEAKY * p; s = fmaf(t4.z, p, s);
            p = fmaf(w4.w, a4.w, b4.w); p = (p >= 0.f) ? p : LEAKY * p; s = fmaf(t4.w, p, s);
        }
        att[h * 40 + n] = expf(s);               // reference: exp without max-shift
    }
    __syncthreads();

    // ---- softmax normalization over n (tiny; 4 threads) ----
    if (tid < NUM_HEADS) {
        float s = 0.f;
        for (int n = 0; n < NWORDS; ++n) s += att[tid * 40 + n];
        float inv = 1.f / s;
        for (int n = 0; n < NWORDS; ++n) att[tid * 40 + n] *= inv;
    }
    __syncthreads();

    // ---- out(16x112) = att(16x40) @ words(40x112) via V_WMMA_F32_16X16X4_F32 ----
    const int wave = tid >> 5;
    const int lane = tid & 31;
    if (wave < 7) {                              // wave-uniform branch: EXEC all-1s inside
        const int tile = wave;                   // 16 output columns per wave
        const int l    = lane & 15;
        const int ksub = (lane >> 4) * 2;        // lanes 0-15: K={k,k+1}; 16-31: K={k+2,k+3}
        v8f c = {0.f, 0.f, 0.f, 0.f, 0.f, 0.f, 0.f, 0.f};
        for (int k = 0; k < 40; k += 4) {
            v2f a, b;
            // A 16x4 fp32 layout (ISA 7.12.2): lane row M=l, two K values per lane half
            a[0] = att[l * 40 + k + ksub];
            a[1] = att[l * 40 + k + ksub + 1];
            // B 4x16 fp32: B[K][N] = words[n=K][d=tile*16+N]; word row K -> xs row K+1
            b[0] = xs[(k + ksub + 1) * DPAD + tile * 16 + l];
            b[1] = xs[(k + ksub + 2) * DPAD + tile * 16 + l];
            c = __builtin_amdgcn_wmma_f32_16x16x4_f32(
                    false, a, false, b, (short)0, c, false, false);
        }
        // D layout: VGPR v holds M=v for lanes 0-15 (N=lane). Heads are rows 0..3.
        const int d = tile * 16 + l;
        if (lane < 16 && d < WORD_DIM) {
            float* og = out + (size_t)ent * (NUM_HEADS * WORD_DIM) + d;
            og[0 * WORD_DIM] = c[0] * flagf;
            og[1 * WORD_DIM] = c[1] * flagf;
            og[2 * WORD_DIM] = c[2] * flagf;
            og[3 * WORD_DIM] = c[3] * flagf;
        }
    }
}

extern "C" void kernel_launch(void* const* d_in, const int* in_sizes, int n_in,
                              void* d_out, int out_size, void* d_ws, size_t ws_size,
                              hipStream_t stream) {
    const float* x     = (const float*)d_in[0];
    const float* w_att = (const float*)d_in[1];
    const float* b_att = (const float*)d_in[2];
    float* out = (float*)d_out;
    (void)in_sizes; (void)n_in; (void)out_size; (void)d_ws; (void)ws_size;
    tse_kernel<<<dim3(BTE), dim3(256), 0, stream>>>(x, w_att, b_att, out);
}